// GMNLayer_84112639525114
// MI455X (gfx1250) — compile-verified
//
#include <hip/hip_runtime.h>

typedef __attribute__((ext_vector_type(2))) float v2f;
typedef __attribute__((ext_vector_type(8))) float v8f;

enum {
  NN = 1024,          // nodes
  NE = 32768,         // edges
  KE = 32,            // nedgeinput / output of W2..W5
  KO = 32,            // nedgeoutput
  DI = 64,            // ninp
  DO = 64,            // nout1
  DN = 64,            // fcnode out dim
  TMPW = 3 * KE + DN, // 160  (concat width)
  HIDW = 8 * KE       // 256  (W6 output width)
};

// Async global->LDS copy of one 16B chunk (ASYNCcnt-tracked, no VGPR data).
__device__ __forceinline__ void async_b128(const float* g, const float* l) {
  const unsigned loff = (unsigned)(uintptr_t)l;  // low 32 bits = LDS offset
  asm volatile("global_load_async_to_lds_b128 %0, %1, off"
               :: "v"(loff), "v"(g) : "memory");
}
__device__ __forceinline__ void async_wait0() {
  asm volatile("s_wait_asynccnt 0x0" ::: "memory");
}

// ---------------------------------------------------------------------------
// Wave-level f32 WMMA GEMM: each wave computes one 16x16 tile of
// C[M,N] = act(A[M,K] @ B)  (B either W[N,K] row-major -> B(k,n)=W[n,k],
// or plain [K,N] row-major). K must be a multiple of 4.
// ---------------------------------------------------------------------------
template <bool TRANSB, bool RELU, bool BIAS>
__global__ __launch_bounds__(256) void k_gemm(const float* __restrict__ A, int lda,
                                              const float* __restrict__ B, int ldb,
                                              const float* __restrict__ bias,
                                              float* __restrict__ C, int ldc,
                                              int M, int N, int K) {
  const int ntN = N >> 4;
  const int tile = blockIdx.x * 8 + (threadIdx.x >> 5);
  if (tile >= (M >> 4) * ntN) return;  // wave-uniform exit (EXEC stays all-ones)
  const int m0 = (tile / ntN) << 4;
  const int n0 = (tile % ntN) << 4;
  const int lane = threadIdx.x & 31;
  const int kh = lane >> 4;   // half-select: K pair 0/1 vs 2/3
  const int lm = lane & 15;
  v8f acc = {};
  for (int k0 = 0; k0 < K; k0 += 4) {
    v2f a, b;
    const float* ar = A + (size_t)(m0 + lm) * lda + k0 + 2 * kh;
    a.x = ar[0]; a.y = ar[1];
    if (TRANSB) {
      const float* br = B + (size_t)(n0 + lm) * ldb + k0 + 2 * kh;
      b.x = br[0]; b.y = br[1];
    } else {
      b.x = B[(size_t)(k0 + 2 * kh) * ldb + n0 + lm];
      b.y = B[(size_t)(k0 + 2 * kh + 1) * ldb + n0 + lm];
    }
    acc = __builtin_amdgcn_wmma_f32_16x16x4_f32(false, a, false, b, (short)0, acc,
                                                false, false);
  }
#pragma unroll
  for (int r = 0; r < 8; ++r) {
    const int m = m0 + r + 8 * kh;
    const int n = n0 + lm;
    float v = acc[r];
    if (RELU) v = fmaxf(v, 0.f);
    if (BIAS) v += bias[n];
    C[(size_t)m * ldc + n] = v;
  }
}

// ---------------------------------------------------------------------------
// Graph structure: dense edge-id map + deterministic CSR by source
// ---------------------------------------------------------------------------
__global__ void k_init(int* __restrict__ map, int* __restrict__ deg) {
  const int i = blockIdx.x * blockDim.x + threadIdx.x;
  if (i < NN * NN) map[i] = -1;
  if (i < NN) deg[i] = 0;
}

__global__ void k_scatter(const int* __restrict__ src, const int* __restrict__ dst,
                          int* __restrict__ map, int* __restrict__ deg) {
  const int e = blockIdx.x * blockDim.x + threadIdx.x;
  if (e >= NE) return;
  map[src[e] * NN + dst[e]] = e;
  atomicAdd(&deg[src[e]], 1);  // order-independent -> deterministic
}

__global__ __launch_bounds__(1024) void k_scan(const int* __restrict__ deg,
                                               int* __restrict__ row_start) {
  __shared__ int s[NN];
  const int t = threadIdx.x;
  s[t] = deg[t];
  __syncthreads();
  for (int off = 1; off < NN; off <<= 1) {
    const int v = (t >= off) ? s[t - off] : 0;
    __syncthreads();
    s[t] += v;
    __syncthreads();
  }
  row_start[t] = s[t] - deg[t];
  if (t == NN - 1) row_start[NN] = s[NN - 1];
}

// Deterministic CSR fill: each node scans edges in order (float-sum order fixed).
__global__ void k_build_csr(const int* __restrict__ src, const int* __restrict__ row_start,
                            int* __restrict__ csr) {
  const int n = blockIdx.x * blockDim.x + threadIdx.x;
  if (n >= NN) return;
  int p = row_start[n];
  for (int e = 0; e < NE; ++e)
    if (src[e] == n) csr[p++] = e;
}

// ---------------------------------------------------------------------------
// Sparse A@B gathered on edges + assembly of tmp = [SP | X2*X3 | diag | mm]
// One wave per edge; lane = k channel. All control uniform per wave.
// ---------------------------------------------------------------------------
__global__ __launch_bounds__(256) void k_sparse_assemble(
    const int* __restrict__ src, const int* __restrict__ dst,
    const int* __restrict__ map, const int* __restrict__ row_start,
    const int* __restrict__ csr, const float* __restrict__ SP,
    const float* __restrict__ X2, const float* __restrict__ X3,
    const float* __restrict__ X4, const float* __restrict__ X5,
    const float* __restrict__ h, float* __restrict__ tmp) {
  const int e = (blockIdx.x * blockDim.x + threadIdx.x) >> 5;
  const int lane = threadIdx.x & 31;
  if (e >= NE) return;
  const int u = src[e], v = dst[e];
  float acc = 0.f;
  const int beg = row_start[u], end = row_start[u + 1];
  for (int i = beg; i < end; ++i) {
    const int e1 = csr[i];          // edge (u, m)
    const int m = dst[e1];
    const int e2 = map[m * NN + v]; // edge (m, v), -1 if absent
    if (e2 >= 0) acc = fmaf(X4[(size_t)e1 * KE + lane], X5[(size_t)e2 * KE + lane], acc);
  }
  float* trow = tmp + (size_t)e * TMPW;
  trow[lane]      = SP[(size_t)e * KE + lane];
  trow[KE + lane] = X2[(size_t)e * KE + lane] * X3[(size_t)e * KE + lane];
  float d0 = 0.f, d1 = 0.f;
  if (u == v) { d0 = h[(size_t)u * DN + 2 * lane]; d1 = h[(size_t)u * DN + 2 * lane + 1]; }
  trow[2 * KE + 2 * lane]     = d0;
  trow[2 * KE + 2 * lane + 1] = d1;
  trow[2 * KE + DN + lane]    = acc;
}

// ---------------------------------------------------------------------------
// Fused edge MLP: edge_attr = relu(tmp @ W6.T) @ W7.T, hidden kept in LDS.
// One block per 32 edges (2 M-tiles): each wave reuses every W6 B-fragment
// across both M-tiles; stage 2 uses 4/8 waves. tmp tile staged via
// global_load_async_to_lds_b128 (ASYNCcnt).
// ---------------------------------------------------------------------------
#define TMP_LD 161  // 161 % 64 odd  -> conflict-free column access
#define HID_LD 257  // 257 % 64 == 1 -> conflict-free column access
__global__ __launch_bounds__(256) void k_edge_mlp(const float* __restrict__ tmp,
                                                  const float* __restrict__ W6,
                                                  const float* __restrict__ W7,
                                                  float* __restrict__ edge_attr) {
  __shared__ float s_tmp[32 * TMP_LD];
  __shared__ float s_hid[32 * HID_LD];
  const int m0 = blockIdx.x * 32;
  const int t = threadIdx.x;

  // async-stage tmp[32,160] -> LDS: 1280 x 16B chunks, 5 per thread
#pragma unroll
  for (int j = 0; j < 5; ++j) {
    const int i = t + j * 256;
    const int r = i / 40, c = (i % 40) * 4;
    async_b128(tmp + (size_t)(m0 + r) * TMPW + c, &s_tmp[r * TMP_LD + c]);
  }
  async_wait0();
  __syncthreads();

  const int wave = t >> 5, lane = t & 31, kh = lane >> 4, lm = lane & 15;

  // Stage 1: wave w -> hidden columns [w*32, w*32+32), both M-tiles
  {
    const int n0 = wave * 32;
    v8f a00 = {}, a01 = {}, a10 = {}, a11 = {};
    for (int k0 = 0; k0 < TMPW; k0 += 4) {
      v2f aA, aB, b0, b1;
      aA.x = s_tmp[lm * TMP_LD + k0 + 2 * kh];
      aA.y = s_tmp[lm * TMP_LD + k0 + 2 * kh + 1];
      aB.x = s_tmp[(16 + lm) * TMP_LD + k0 + 2 * kh];
      aB.y = s_tmp[(16 + lm) * TMP_LD + k0 + 2 * kh + 1];
      const float* w0 = W6 + (size_t)(n0 + lm) * TMPW + k0 + 2 * kh;
      const float* w1 = W6 + (size_t)(n0 + 16 + lm) * TMPW + k0 + 2 * kh;
      b0.x = w0[0]; b0.y = w0[1];
      b1.x = w1[0]; b1.y = w1[1];
      a00 = __builtin_amdgcn_wmma_f32_16x16x4_f32(false, aA, false, b0, (short)0, a00, false, false);
      a01 = __builtin_amdgcn_wmma_f32_16x16x4_f32(false, aA, false, b1, (short)0, a01, false, false);
      a10 = __builtin_amdgcn_wmma_f32_16x16x4_f32(false, aB, false, b0, (short)0, a10, false, false);
      a11 = __builtin_amdgcn_wmma_f32_16x16x4_f32(false, aB, false, b1, (short)0, a11, false, false);
    }
#pragma unroll
    for (int r = 0; r < 8; ++r) {
      const int mr = r + 8 * kh;
      s_hid[mr * HID_LD + n0 + lm]             = fmaxf(a00[r], 0.f);
      s_hid[mr * HID_LD + n0 + 16 + lm]        = fmaxf(a01[r], 0.f);
      s_hid[(16 + mr) * HID_LD + n0 + lm]      = fmaxf(a10[r], 0.f);
      s_hid[(16 + mr) * HID_LD + n0 + 16 + lm] = fmaxf(a11[r], 0.f);
    }
  }
  __syncthreads();

  // Stage 2: waves 0..3 -> edge_attr tiles (mt = w>>1, nt = w&1), K = 256
  if (wave < 4) {
    const int mt = wave >> 1;
    const int n0 = (wave & 1) * 16;
    v8f acc = {};
    for (int k0 = 0; k0 < HIDW; k0 += 4) {
      v2f a, b;
      a.x = s_hid[(mt * 16 + lm) * HID_LD + k0 + 2 * kh];
      a.y = s_hid[(mt * 16 + lm) * HID_LD + k0 + 2 * kh + 1];
      const float* wr = W7 + (size_t)(n0 + lm) * HIDW + k0 + 2 * kh;
      b.x = wr[0]; b.y = wr[1];
      acc = __builtin_amdgcn_wmma_f32_16x16x4_f32(false, a, false, b, (short)0, acc, false, false);
    }
#pragma unroll
    for (int r = 0; r < 8; ++r)
      edge_attr[(size_t)(m0 + mt * 16 + r + 8 * kh) * KO + n0 + lm] = acc[r];
  }
}

// ---------------------------------------------------------------------------
// SpectConv aggregation: agg[n,k,c] = sum_{e: src=n} edge_attr[e,k] * x[dst_e,c]
// One block per node; double-buffered async global->LDS prefetch of the next
// edge's (edge_attr row, x[dst] row) while FMA-ing the current one.
// ---------------------------------------------------------------------------
__global__ __launch_bounds__(256) void k_spectconv(const int* __restrict__ row_start,
                                                   const int* __restrict__ csr,
                                                   const int* __restrict__ dst,
                                                   const float* __restrict__ edge_attr,
                                                   const float* __restrict__ x,
                                                   float* __restrict__ agg) {
  __shared__ float ea[2][KO];
  __shared__ float xr[2][DI];
  const int n = blockIdx.x;
  const int t = threadIdx.x;
  const int k = t & 31;
  const int cg = t >> 5;  // 0..7
  float acc[8] = {0.f, 0.f, 0.f, 0.f, 0.f, 0.f, 0.f, 0.f};
  const int beg = row_start[n], end = row_start[n + 1];

  // prefetch helper: 24 x 16B chunks (8 for edge_attr row, 16 for x[dst] row)
  auto issue = [&](int e, int buf) {
    if (t < 8) {
      async_b128(edge_attr + (size_t)e * KO + t * 4, &ea[buf][t * 4]);
    } else if (t < 24) {
      const int d = dst[e];
      async_b128(x + (size_t)d * DI + (t - 8) * 4, &xr[buf][(t - 8) * 4]);
    }
  };

  if (beg < end) issue(csr[beg], 0);
  async_wait0();
  __syncthreads();

  for (int i = beg; i < end; ++i) {
    const int cur = (i - beg) & 1;
    if (i + 1 < end) issue(csr[i + 1], cur ^ 1);   // prefetch next edge
    const float a = ea[cur][k];
#pragma unroll
    for (int j = 0; j < 8; ++j) acc[j] = fmaf(a, xr[cur][cg * 8 + j], acc[j]);
    async_wait0();
    __syncthreads();
  }
#pragma unroll
  for (int j = 0; j < 8; ++j)
    agg[((size_t)n * KO + k) * DI + cg * 8 + j] = acc[j];
}

// ---------------------------------------------------------------------------
extern "C" void kernel_launch(void* const* d_in, const int* in_sizes, int n_in,
                              void* d_out, int out_size, void* d_ws, size_t ws_size,
                              hipStream_t stream) {
  const float* x      = (const float*)d_in[0];
  const int*   ei     = (const int*)d_in[1];
  const float* SP     = (const float*)d_in[2];
  const float* Wnode  = (const float*)d_in[4];
  const float* W2     = (const float*)d_in[5];
  const float* W3     = (const float*)d_in[6];
  const float* W4     = (const float*)d_in[7];
  const float* W5     = (const float*)d_in[8];
  const float* W6     = (const float*)d_in[9];
  const float* W7     = (const float*)d_in[10];
  const float* conv_w = (const float*)d_in[11];
  const float* conv_b = (const float*)d_in[12];
  const int* src = ei;
  const int* dst = ei + NE;

  float* x_out     = (float*)d_out;            // [NN, DO]
  float* edge_attr = x_out + (size_t)NN * DO;  // [NE, KO]

  // workspace carve-out (256B aligned)
  char* ws = (char*)d_ws;
  size_t off = 0;
  auto carve = [&](size_t bytes) {
    void* p = ws + off;
    off += (bytes + 255) & ~(size_t)255;
    return p;
  };
  int*   map       = (int*)carve((size_t)NN * NN * 4);
  int*   deg       = (int*)carve((size_t)NN * 4);
  int*   row_start = (int*)carve((size_t)(NN + 1) * 4);
  int*   csr       = (int*)carve((size_t)NE * 4);
  float* h         = (float*)carve((size_t)NN * DN * 4);
  float* X2        = (float*)carve((size_t)NE * KE * 4);
  float* X3        = (float*)carve((size_t)NE * KE * 4);
  float* X4        = (float*)carve((size_t)NE * KE * 4);
  float* X5        = (float*)carve((size_t)NE * KE * 4);
  float* tmp       = (float*)carve((size_t)NE * TMPW * 4);
  float* agg       = (float*)carve((size_t)NN * KO * DI * 4);
  (void)ws_size; (void)in_sizes; (void)n_in; (void)out_size;

  // 1) graph structure
  k_init<<<(NN * NN) / 256, 256, 0, stream>>>(map, deg);
  k_scatter<<<NE / 256, 256, 0, stream>>>(src, dst, map, deg);
  k_scan<<<1, 1024, 0, stream>>>(deg, row_start);
  k_build_csr<<<NN / 256, 256, 0, stream>>>(src, row_start, csr);

  // 2) dense GEMMs (f32 WMMA): h = x@Wnode.T ; X{2..5} = SP@W{2..5}.T
  k_gemm<true, false, false><<<32, 256, 0, stream>>>(x, DI, Wnode, DI, nullptr,
                                                     h, DN, NN, DN, DI);
  k_gemm<true, false, false><<<512, 256, 0, stream>>>(SP, KE, W2, KE, nullptr,
                                                      X2, KE, NE, KE, KE);
  k_gemm<true, false, false><<<512, 256, 0, stream>>>(SP, KE, W3, KE, nullptr,
                                                      X3, KE, NE, KE, KE);
  k_gemm<true, false, false><<<512, 256, 0, stream>>>(SP, KE, W4, KE, nullptr,
                                                      X4, KE, NE, KE, KE);
  k_gemm<true, false, false><<<512, 256, 0, stream>>>(SP, KE, W5, KE, nullptr,
                                                      X5, KE, NE, KE, KE);

  // 3) sparse matmul gather + tmp assembly (one wave / edge)
  k_sparse_assemble<<<NE / 8, 256, 0, stream>>>(src, dst, map, row_start, csr,
                                                SP, X2, X3, X4, X5, h, tmp);

  // 4) fused edge MLP (W6 relu + W7), hidden lives in LDS, async-staged tmp
  k_edge_mlp<<<NE / 32, 256, 0, stream>>>(tmp, W6, W7, edge_attr);

  // 5) SpectConv aggregation (async double-buffered gather) + final contraction
  k_spectconv<<<NN, 256, 0, stream>>>(row_start, csr, dst, edge_attr, x, agg);
  k_gemm<false, false, true><<<32, 256, 0, stream>>>(agg, KO * DI, conv_w, DO, conv_b,
                                                     x_out, DO, NN, DO, KO * DI);
}